// CategorySpecificMLP_4741643895622
// MI455X (gfx1250) — compile-verified
//
#include <hip/hip_runtime.h>
#include <stdint.h>

typedef __attribute__((ext_vector_type(2))) float v2f;
typedef __attribute__((ext_vector_type(8))) float v8f;

typedef __attribute__((address_space(1))) int glob_int;   // builtin wants int* per hipcc diagnostic
typedef __attribute__((address_space(3))) int lds_int;

// Problem constants (fixed by the reference).
constexpr int N_ = 8192;
constexpr int C_ = 100;
constexpr int D_ = 128;   // input dim  (K of layer 1)
constexpr int H_ = 128;   // hidden dim (N of layer 1, K of layer 2)
constexpr int O_ = 64;    // output dim (N of layer 2)

// LDS layout (padded strides to dodge bank conflicts; even strides keep b64 alignment)
constexpr int W1T_S = 130;          // W1^T row stride in floats: [H][D] padded
constexpr int W2T_S = 130;          // W2^T row stride in floats: [O][H] padded
constexpr int HS_S  = 132;          // per-wave h scratch row stride: [16][H] padded
constexpr int WAVES = 4;
constexpr int BLK   = 32 * WAVES;
constexpr int GRIDY = 2;
constexpr int SMEM_FLOATS = H_ * W1T_S + O_ * W2T_S + H_ + O_ + WAVES * 16 * HS_S;

#if defined(__gfx1250__) && __has_builtin(__builtin_amdgcn_global_load_async_to_lds_b32)
#define ASYNC_STAGE 1
#endif

__device__ __forceinline__ void stage_f32(const float* g, float* l) {
#ifdef ASYNC_STAGE
    // CDNA5 GLOBAL_LOAD_ASYNC_TO_LDS_B32: memory -> LDS without a VGPR round
    // trip; per-lane LDS destination address lets us keep the padded transpose.
    // Low 32 bits of a generic LDS address are the LDS byte offset.
    __builtin_amdgcn_global_load_async_to_lds_b32(
        (glob_int*)(uintptr_t)g,
        (lds_int*)(uint32_t)(uintptr_t)l, 0, 0);
#else
    *l = *g;
#endif
}

__device__ __forceinline__ void stage_wait() {
#ifdef ASYNC_STAGE
#if __has_builtin(__builtin_amdgcn_s_wait_asynccnt)
    __builtin_amdgcn_s_wait_asynccnt(0);
#else
    asm volatile("s_wait_asynccnt 0x0" ::: "memory");
#endif
#endif
}

// ---------------- bucketing prologue ----------------

__global__ void cat_zero_kernel(int* counts) {
    int i = blockIdx.x * blockDim.x + threadIdx.x;
    if (i < C_) counts[i] = 0;
}

__global__ void cat_hist_kernel(const int* __restrict__ cat, int* __restrict__ counts) {
    int i = blockIdx.x * blockDim.x + threadIdx.x;
    if (i < N_) atomicAdd(&counts[cat[i]], 1);
}

__global__ void cat_scan_kernel(const int* __restrict__ counts,
                                int* __restrict__ offsets, int* __restrict__ cursor) {
    if (threadIdx.x == 0) {
        int run = 0;
        for (int c = 0; c < C_; ++c) {
            offsets[c] = run;
            cursor[c]  = run;
            run += counts[c];
        }
        offsets[C_] = run;
    }
}

__global__ void cat_scatter_kernel(const int* __restrict__ cat,
                                   int* __restrict__ cursor, int* __restrict__ perm) {
    int i = blockIdx.x * blockDim.x + threadIdx.x;
    if (i < N_) {
        int p = atomicAdd(&cursor[cat[i]], 1);
        perm[p] = i;
    }
}

// ---------------- main fused MLP kernel ----------------
// grid = (C, GRIDY), block = 128 (4 wave32s). One category per blockIdx.x.
// Each wave processes 16-token tiles of its category with f32 WMMA (16x16x4).

__global__ __launch_bounds__(BLK) void mlp_wmma_kernel(
    const float* __restrict__ x,    const int* __restrict__ perm,
    const int*  __restrict__ offsets,
    const float* __restrict__ W1,   const float* __restrict__ b1,
    const float* __restrict__ W2,   const float* __restrict__ b2,
    float* __restrict__ out)
{
    __shared__ float smem[SMEM_FLOATS];   // ~134 KB; CDNA5 WGP has 320 KB LDS
    float* w1t   = smem;                          // [H][D] padded (W1 transposed)
    float* w2t   = w1t + H_ * W1T_S;              // [O][H] padded (W2 transposed)
    float* b1s   = w2t + O_ * W2T_S;              // [H]
    float* b2s   = b1s + H_;                      // [O]
    float* hsAll = b2s + O_;                      // WAVES * [16][H] padded

    const int c   = blockIdx.x;
    const int tid = threadIdx.x;

    // Stage this category's weights into LDS (async global->LDS when available;
    // coalesced reads, padded-stride scatter -> conflict-light).
    const float* W1c = W1 + (size_t)c * (D_ * H_);   // row-major [D][H]
    for (int i = tid; i < D_ * H_; i += BLK) {
        int d = i >> 7, h = i & (H_ - 1);
        stage_f32(&W1c[i], &w1t[h * W1T_S + d]);
    }
    const float* W2c = W2 + (size_t)c * (H_ * O_);   // row-major [H][O]
    for (int i = tid; i < H_ * O_; i += BLK) {
        int h = i >> 6, o = i & (O_ - 1);
        stage_f32(&W2c[i], &w2t[o * W2T_S + h]);
    }
    for (int i = tid; i < H_; i += BLK) stage_f32(&b1[(size_t)c * H_ + i], &b1s[i]);
    for (int i = tid; i < O_; i += BLK) stage_f32(&b2[(size_t)c * O_ + i], &b2s[i]);
    stage_wait();
    __syncthreads();

    const int lane = tid & 31;
    const int wave = tid >> 5;
    const int row  = lane & 15;            // A-row / B-col / C-col index for this lane
    const int kb   = (lane >> 4) << 1;     // K sub-offset: 0 for lanes 0-15, 2 for 16-31
    const int mB   = (lane >> 4) << 3;     // C-layout row base: 0 or 8

    const int begin  = offsets[c];
    const int end    = offsets[c + 1];
    const int ntiles = (end - begin + 15) >> 4;
    float* hs = hsAll + wave * (16 * HS_S);

    const v2f zero2 = {};
    const v8f zero8 = {};

    for (int t = blockIdx.y * WAVES + wave; t < ntiles; t += GRIDY * WAVES) {
        const int base = begin + t * 16;

        // Gather my A row (token row of x) into 32 float2 fragments: whole 16x128
        // tile lives in 64 VGPRs/wave and is reused across all 8 N-tiles.
        int myTok = -1;
        { int idx = base + row; if (idx < end) myTok = perm[idx]; }
        v2f afrag[32];
        if (myTok >= 0) {
            const float* xr = x + (size_t)myTok * D_;
            #pragma unroll
            for (int s = 0; s < 32; ++s)
                afrag[s] = *(const v2f*)(xr + s * 4 + kb);
        } else {
            #pragma unroll
            for (int s = 0; s < 32; ++s) afrag[s] = zero2;
        }

        // ---- Layer 1: h = relu(A @ W1 + b1), 8 N-tiles x 32 K-steps ----
        // Two independent accumulator chains per N-tile (+unroll 2) for WMMA ILP:
        // occupancy is ~1 wave/SIMD, so in-wave ILP is the only latency hiding.
        #pragma unroll 2
        for (int nt = 0; nt < 8; ++nt) {
            const float bv = b1s[nt * 16 + row];         // bias depends only on column
            v8f accA = {bv, bv, bv, bv, bv, bv, bv, bv}; // bias pre-loaded into chain A
            v8f accB = zero8;
            const float* bcol = &w1t[(nt * 16 + row) * W1T_S];
            #pragma unroll
            for (int s = 0; s < 32; s += 2) {
                v2f bfA = *(const v2f*)(bcol + s * 4 + kb);        // ds_load_2addr_b64
                v2f bfB = *(const v2f*)(bcol + (s + 1) * 4 + kb);
                accA = __builtin_amdgcn_wmma_f32_16x16x4_f32(
                        false, afrag[s],     false, bfA, (short)0, accA, false, false);
                accB = __builtin_amdgcn_wmma_f32_16x16x4_f32(
                        false, afrag[s + 1], false, bfB, (short)0, accB, false, false);
            }
            // ReLU + park in per-wave LDS scratch (C layout -> memory layout)
            #pragma unroll
            for (int v = 0; v < 8; ++v)
                hs[(mB + v) * HS_S + nt * 16 + row] = fmaxf(accA[v] + accB[v], 0.0f);
        }

        // Re-shape h from C layout to A-fragment layout (LDS transpose bounce;
        // same-wave LDS ops are in-order, compiler inserts the dscnt wait).
        v2f afrag2[32];
        #pragma unroll
        for (int s = 0; s < 32; ++s)
            afrag2[s] = *(const v2f*)&hs[row * HS_S + s * 4 + kb];

        // ---- Layer 2: out = h @ W2 + b2, 4 N-tiles x 32 K-steps ----
        #pragma unroll 2
        for (int nt = 0; nt < 4; ++nt) {
            const float bv = b2s[nt * 16 + row];
            v8f accA = {bv, bv, bv, bv, bv, bv, bv, bv};
            v8f accB = zero8;
            const float* bcol = &w2t[(nt * 16 + row) * W2T_S];
            #pragma unroll
            for (int s = 0; s < 32; s += 2) {
                v2f bfA = *(const v2f*)(bcol + s * 4 + kb);
                v2f bfB = *(const v2f*)(bcol + (s + 1) * 4 + kb);
                accA = __builtin_amdgcn_wmma_f32_16x16x4_f32(
                        false, afrag2[s],     false, bfA, (short)0, accA, false, false);
                accB = __builtin_amdgcn_wmma_f32_16x16x4_f32(
                        false, afrag2[s + 1], false, bfB, (short)0, accB, false, false);
            }
            #pragma unroll
            for (int v = 0; v < 8; ++v) {
                int idx = base + mB + v;
                if (idx < end)
                    out[(size_t)perm[idx] * O_ + nt * 16 + row] = accA[v] + accB[v];
            }
        }
    }
}

// ---------------- launcher ----------------

extern "C" void kernel_launch(void* const* d_in, const int* in_sizes, int n_in,
                              void* d_out, int out_size, void* d_ws, size_t ws_size,
                              hipStream_t stream) {
    const float* x   = (const float*)d_in[0];
    const int*   cat = (const int*)  d_in[1];
    const float* W1  = (const float*)d_in[2];
    const float* b1  = (const float*)d_in[3];
    const float* W2  = (const float*)d_in[4];
    const float* b2  = (const float*)d_in[5];
    float*       out = (float*)d_out;

    // Workspace layout (ints): counts[C] | offsets[C+1] | cursor[C] | perm[N]
    int* counts  = (int*)d_ws;
    int* offsets = counts + C_;
    int* cursor  = offsets + C_ + 1;
    int* perm    = cursor + C_;

    cat_zero_kernel<<<(C_ + 127) / 128, 128, 0, stream>>>(counts);
    cat_hist_kernel<<<(N_ + 255) / 256, 256, 0, stream>>>(cat, counts);
    cat_scan_kernel<<<1, 32, 0, stream>>>(counts, offsets, cursor);
    cat_scatter_kernel<<<(N_ + 255) / 256, 256, 0, stream>>>(cat, cursor, perm);

    dim3 grid(C_, GRIDY);
    mlp_wmma_kernel<<<grid, BLK, 0, stream>>>(x, perm, offsets, W1, b1, W2, b2, out);
}